// GMRESSolver_68942815036041
// MI455X (gfx1250) — compile-verified
//
#include <hip/hip_runtime.h>
#include <math.h>

// ---------------------------------------------------------------------------
// GMRES(16) x 4 restart cycles for -Lap(periodic 5pt) + k^2 on 1024x1024 f32.
// MI455X (gfx1250, wave32). Bandwidth-bound field kernels use float4
// (global_load_b128) streaming, fused CGS2 passes, deterministic 2-level
// reductions with a single-barrier 17-way block reduce. WMMA f32 16x16x4
// computes the Gram matrix G = H^T H of the Hessenberg least-squares solve.
// ---------------------------------------------------------------------------

#define N       1024
#define NN      (N * N)
#define NN4     (NN / 4)
#define RESTART 16
#define NCYC    4
#define K2F     0.3f
#define EPSF    1e-12f
#define NB      1024      // blocks for field kernels / partial sums
#define NT      256       // threads per block (8 waves on CDNA5)
#define NWAVES  (NT / 32)

static_assert((size_t)NB * NT == NN4, "one float4 quad per thread");

typedef __attribute__((ext_vector_type(2))) float v2f;
typedef __attribute__((ext_vector_type(8))) float v8f;

// ---- workspace layout (float offsets) -------------------------------------
// V[17][NN] | w[NN] | partial[NB*32] | scal[16] | h1[32] | h2[32] | Hm[288] | y[16]
static constexpr size_t OFF_V  = 0;
static constexpr size_t OFF_W  = (size_t)17 * NN;
static constexpr size_t OFF_P  = OFF_W + NN;
static constexpr size_t OFF_SC = OFF_P + (size_t)NB * 32;
static constexpr size_t OFF_H1 = OFF_SC + 16;
static constexpr size_t OFF_H2 = OFF_H1 + 32;
static constexpr size_t OFF_HM = OFF_H2 + 32;     // 17*16 = 272, padded to 288
static constexpr size_t OFF_Y  = OFF_HM + 288;
static constexpr size_t WS_FLOATS = OFF_Y + 16;   // ~18.03M floats (~72.2 MB)

// scal[] slots
#define SC_BETA   0
#define SC_INVB   1
#define SC_HN     2
#define SC_INVHN  3

// ---- reduction helpers ------------------------------------------------------
__device__ __forceinline__ float waveReduce(float v) {
#pragma unroll
  for (int o = 16; o > 0; o >>= 1) v += __shfl_down(v, o, 32);
  return v;
}

// generic block reduction (valid in thread 0); used by the small 1-block kernels
__device__ float blockReduceSum(float v) {
  __shared__ float smem[32];
  const int lane = threadIdx.x & 31;
  const int wid  = threadIdx.x >> 5;
  v = waveReduce(v);
  __syncthreads();                       // protect smem across repeated calls
  if (lane == 0) smem[wid] = v;
  __syncthreads();
  const int nw = (blockDim.x + 31) >> 5;
  v = (threadIdx.x < (unsigned)nw) ? smem[threadIdx.x] : 0.f;
  if (wid == 0) v = waveReduce(v);
  return v;
}

// 17-way block reduction with ONE barrier; threads 0..jc-1 write the block's
// partials directly. Call once at end of kernel.
__device__ __forceinline__ void blockReduceVec17(const float (&acc)[17], int jc,
                                                 float* __restrict__ out) {
  __shared__ float smem[NWAVES][18];
  const int lane = threadIdx.x & 31;
  const int wid  = threadIdx.x >> 5;
#pragma unroll
  for (int k = 0; k < 17; ++k)
    if (k < jc) {                        // uniform: jc is an SGPR
      const float v = waveReduce(acc[k]);
      if (lane == 0) smem[wid][k] = v;
    }
  __syncthreads();
  if ((int)threadIdx.x < jc) {
    float s = 0.f;
#pragma unroll
    for (int wv = 0; wv < NWAVES; ++wv) s += smem[wv][threadIdx.x];
    out[threadIdx.x] = s;
  }
}

// ---- stencil core (vectorized) ---------------------------------------------
// Quad i4 covers elements [iy, 4*qx .. 4*qx+3]; 256 quads per row.
__device__ __forceinline__ float4 applyA4(const float* __restrict__ u, int i4) {
  const float4* __restrict__ u4 = (const float4*)u;
  const int iy = i4 >> 8;
  const int ix = (i4 & 255) << 2;
  const int rowbase = iy * N;
  const float4 c  = u4[i4];
  const float4 ym = u4[(i4 - (N / 4) + NN4) & (NN4 - 1)];  // periodic rows
  const float4 yp = u4[(i4 + (N / 4)) & (NN4 - 1)];
  const float left  = u[rowbase + ((ix + N - 1) & (N - 1))];
  const float right = u[rowbase + ((ix + 4) & (N - 1))];
  float4 o;
  o.x = -(left + c.y + ym.x + yp.x - 4.f * c.x) + K2F * c.x;
  o.y = -(c.x  + c.z + ym.y + yp.y - 4.f * c.y) + K2F * c.y;
  o.z = -(c.y  + c.w + ym.z + yp.z - 4.f * c.z) + K2F * c.z;
  o.w = -(c.z  + right + ym.w + yp.w - 4.f * c.w) + K2F * c.w;
  return o;
}

// ---- field kernels (one quad per thread, b128 traffic) ----------------------
// r = b - A(x) into V0, partial ||r||^2 per block
__global__ __launch_bounds__(NT) void k_residual(const float* __restrict__ x,
                                                 const float* __restrict__ b,
                                                 float* __restrict__ V0,
                                                 float* __restrict__ partial) {
  const int i4 = blockIdx.x * NT + threadIdx.x;
  const float4 a = applyA4(x, i4);
  const float4 bv = ((const float4*)b)[i4];
  float4 r;
  r.x = bv.x - a.x; r.y = bv.y - a.y; r.z = bv.z - a.z; r.w = bv.w - a.w;
  ((float4*)V0)[i4] = r;
  const float acc = r.x * r.x + r.y * r.y + r.z * r.z + r.w * r.w;
  const float s = blockReduceSum(acc);
  if (threadIdx.x == 0) partial[blockIdx.x * 32] = s;
}

__global__ __launch_bounds__(NT) void k_stencil(const float* __restrict__ vj,
                                                float* __restrict__ w) {
  const int i4 = blockIdx.x * NT + threadIdx.x;
  ((float4*)w)[i4] = applyA4(vj, i4);
}

// dst = src * scal[idx]   (in-place allowed)
__global__ __launch_bounds__(NT) void k_scale(float* __restrict__ dst,
                                              const float* __restrict__ src,
                                              const float* __restrict__ scal,
                                              int idx) {
  const float f = scal[idx];
  const int i4 = blockIdx.x * NT + threadIdx.x;
  const float4 v = ((const float4*)src)[i4];
  float4 o;
  o.x = v.x * f; o.y = v.y * f; o.z = v.z * f; o.w = v.w * f;
  ((float4*)dst)[i4] = o;
}

// partial h[k] = sum_i V[k][i]*w[i], k < jc  (first CGS pass)
__global__ __launch_bounds__(NT) void k_dot_partial(const float* __restrict__ V,
                                                    const float* __restrict__ w,
                                                    int jc,
                                                    float* __restrict__ partial) {
  const int i4 = blockIdx.x * NT + threadIdx.x;
  const float4* __restrict__ V4 = (const float4*)V;
  const float4 wv = ((const float4*)w)[i4];
  float acc[17];
#pragma unroll
  for (int k = 0; k < 17; ++k) {
    acc[k] = 0.f;
    if (k < jc) {
      const float4 v = V4[(size_t)k * NN4 + i4];
      acc[k] = v.x * wv.x + v.y * wv.y + v.z * wv.z + v.w * wv.w;
    }
  }
  blockReduceVec17(acc, jc, partial + (size_t)blockIdx.x * 32);
}

// fused: w -= sum_k h[k]*V[k]; then either partial dots V.w (DO_DOT=1, feeds
// second CGS pass) or partial ||w||^2 (DO_DOT=0, feeds the norm). V is read
// exactly once for both the AXPY and the reduction.
template <int DO_DOT>
__global__ __launch_bounds__(NT) void k_sub_fused(float* __restrict__ w,
                                                  const float* __restrict__ V,
                                                  const float* __restrict__ h,
                                                  int jc,
                                                  float* __restrict__ partial) {
  const int i4 = blockIdx.x * NT + threadIdx.x;
  const float4* __restrict__ V4 = (const float4*)V;
  float hk[17];
#pragma unroll
  for (int k = 0; k < 17; ++k) hk[k] = (k < jc) ? h[k] : 0.f;

  float4 vv[17];
  float4 wn = ((const float4*)w)[i4];
#pragma unroll
  for (int k = 0; k < 17; ++k)
    if (k < jc) {
      const float4 v = V4[(size_t)k * NN4 + i4];
      if (DO_DOT) vv[k] = v;
      wn.x -= hk[k] * v.x; wn.y -= hk[k] * v.y;
      wn.z -= hk[k] * v.z; wn.w -= hk[k] * v.w;
    }
  ((float4*)w)[i4] = wn;

  if (DO_DOT) {
    float acc[17];
#pragma unroll
    for (int k = 0; k < 17; ++k) {
      acc[k] = 0.f;
      if (k < jc)
        acc[k] = vv[k].x * wn.x + vv[k].y * wn.y + vv[k].z * wn.z + vv[k].w * wn.w;
    }
    blockReduceVec17(acc, jc, partial + (size_t)blockIdx.x * 32);
  } else {
    const float nacc = wn.x * wn.x + wn.y * wn.y + wn.z * wn.z + wn.w * wn.w;
    const float s = blockReduceSum(nacc);
    if (threadIdx.x == 0) partial[blockIdx.x * 32] = s;
  }
}

// x += sum_{k<16} y[k] * V[k]
__global__ __launch_bounds__(NT) void k_update(float* __restrict__ x,
                                               const float* __restrict__ V,
                                               const float* __restrict__ y) {
  const int i4 = blockIdx.x * NT + threadIdx.x;
  const float4* __restrict__ V4 = (const float4*)V;
  float yk[RESTART];
#pragma unroll
  for (int k = 0; k < RESTART; ++k) yk[k] = y[k];
  float4 s = ((float4*)x)[i4];
#pragma unroll
  for (int k = 0; k < RESTART; ++k) {
    const float4 v = V4[(size_t)k * NN4 + i4];
    s.x += yk[k] * v.x; s.y += yk[k] * v.y;
    s.z += yk[k] * v.z; s.w += yk[k] * v.w;
  }
  ((float4*)x)[i4] = s;
}

// ---- small reductions / bookkeeping ----------------------------------------
__global__ __launch_bounds__(NT) void k_zero(float* __restrict__ p, int n) {
  for (int i = threadIdx.x; i < n; i += NT) p[i] = 0.f;
}

// scal[idx]=sqrt(sum partial[b*32]); scal[idx+1]=1/(sqrt+eps)
__global__ __launch_bounds__(NT) void k_reduce_scalar(const float* __restrict__ partial,
                                                      float* __restrict__ scal,
                                                      int idx) {
  float s = 0.f;
  for (int b = threadIdx.x; b < NB; b += NT) s += partial[b * 32];
  s = blockReduceSum(s);
  if (threadIdx.x == 0) {
    const float r = sqrtf(s);
    scal[idx] = r;
    scal[idx + 1] = 1.f / (r + EPSF);
  }
}

__global__ __launch_bounds__(NT) void k_reduce_dots(const float* __restrict__ partial,
                                                    int jc, float* __restrict__ out) {
  for (int k = 0; k < jc; ++k) {
    float s = 0.f;
    for (int b = threadIdx.x; b < NB; b += NT) s += partial[b * 32 + k];
    s = blockReduceSum(s);
    if (threadIdx.x == 0) out[k] = s;
  }
}

// hn = sqrt(sum partial); scal[SC_HN]=hn, scal[SC_INVHN]=1/(hn+eps);
// Hm[:,j] = h1+h2 (k<=j), Hm[j+1,j] = hn
__global__ __launch_bounds__(NT) void k_norm_writeH(const float* __restrict__ partial,
                                                    const float* __restrict__ h1,
                                                    const float* __restrict__ h2,
                                                    float* __restrict__ Hm,
                                                    float* __restrict__ scal,
                                                    int j) {
  float s = 0.f;
  for (int b = threadIdx.x; b < NB; b += NT) s += partial[b * 32];
  s = blockReduceSum(s);
  __shared__ float hn_s;
  if (threadIdx.x == 0) {
    const float hn = sqrtf(s);
    scal[SC_HN] = hn;
    scal[SC_INVHN] = 1.f / (hn + EPSF);
    hn_s = hn;
  }
  __syncthreads();
  if ((int)threadIdx.x <= j)       Hm[threadIdx.x * 16 + j] = h1[threadIdx.x] + h2[threadIdx.x];
  if ((int)threadIdx.x == j + 1)   Hm[threadIdx.x * 16 + j] = hn_s;
}

// ---- Hessenberg normal-equations solve (one wave32, WMMA for G = H^T H) ----
// G(16x16) = sum_{c=0..4} A_c(16x4) x B_c(4x16), K padded 17->20 with zeros.
// A_c[m][kk] = Hm[4c+kk][m], B_c[kk][n] = Hm[4c+kk][n]  ->  A == B (Gram).
// f32 A 16x4 VGPR layout: lanes 0-15 hold K={0,1}, lanes 16-31 hold K={2,3}.
__global__ __launch_bounds__(32) void k_solve(const float* __restrict__ Hm,
                                              const float* __restrict__ scal,
                                              float* __restrict__ y) {
  __shared__ float M[16][18];   // 16x16 G | rhs | pad
  __shared__ float ys[16];
  const int lane = threadIdx.x;
  const int half = lane >> 4;   // 0: K pair {0,1}, 1: K pair {2,3}
  const int mn   = lane & 15;   // row of A == col of B

  v8f G = {};
#pragma unroll
  for (int c = 0; c < 5; ++c) {
    const int k0 = 4 * c + 2 * half;
    v2f a;
    a.x = (k0     < 17) ? Hm[k0 * 16 + mn]       : 0.f;
    a.y = (k0 + 1 < 17) ? Hm[(k0 + 1) * 16 + mn] : 0.f;
    // D = A*B + C ; A == B gives the Gram matrix chunk-accumulated over K.
    G = __builtin_amdgcn_wmma_f32_16x16x4_f32(
        /*neg_a=*/false, a, /*neg_b=*/false, a,
        /*c_mod=*/(short)0, G, /*reuse_a=*/false, /*reuse_b=*/false);
  }
  // C/D layout: VGPR r -> row (r + 8*half), column mn
#pragma unroll
  for (int r = 0; r < 8; ++r) M[r + 8 * half][mn] = G[r];
  __syncthreads();

  if (lane == 0) {
    const float beta = scal[SC_BETA];
    for (int r = 0; r < 16; ++r) {
      M[r][r] += EPSF;                 // + eps*I
      M[r][16] = beta * Hm[r];         // rhs = beta * H^T e1 = beta * Hm[0][:]
    }
    // SPD + eps: plain Gaussian elimination, then back-substitution
    for (int p = 0; p < 16; ++p) {
      const float inv = 1.f / M[p][p];
      for (int r = p + 1; r < 16; ++r) {
        const float f = M[r][p] * inv;
        for (int c = p; c <= 16; ++c) M[r][c] -= f * M[p][c];
      }
    }
    for (int r = 15; r >= 0; --r) {
      float s = M[r][16];
      for (int c = r + 1; c < 16; ++c) s -= M[r][c] * ys[c];
      ys[r] = s / M[r][r];
    }
    for (int r = 0; r < 16; ++r) y[r] = ys[r];
  }
}

// ---------------------------------------------------------------------------
extern "C" void kernel_launch(void* const* d_in, const int* in_sizes, int n_in,
                              void* d_out, int out_size, void* d_ws, size_t ws_size,
                              hipStream_t stream) {
  (void)in_sizes; (void)n_in; (void)out_size;
  const float* src   = (const float*)d_in[0];   // source (b)
  const float* guess = (const float*)d_in[1];   // initial x (zeros)
  float* x  = (float*)d_out;                    // solution lives in d_out
  float* ws = (float*)d_ws;                     // needs >= WS_FLOATS*4 bytes

  float* V       = ws + OFF_V;      // 17 basis fields
  float* w       = ws + OFF_W;
  float* partial = ws + OFF_P;
  float* scal    = ws + OFF_SC;
  float* h1      = ws + OFF_H1;
  float* h2      = ws + OFF_H2;
  float* Hm      = ws + OFF_HM;     // row-major [17][16]
  float* y       = ws + OFF_Y;

  if (ws_size < WS_FLOATS * sizeof(float)) return;  // insufficient scratch

  // x = guess
  hipMemcpyAsync(x, guess, (size_t)NN * sizeof(float),
                 hipMemcpyDeviceToDevice, stream);

  for (int cyc = 0; cyc < NCYC; ++cyc) {
    // zero h1|h2|Hm|y region (fixed-shape; keeps unread H entries at 0)
    k_zero<<<1, NT, 0, stream>>>(h1, (int)(WS_FLOATS - OFF_H1));

    // r = b - A(x); beta; V0 = r/(beta+eps)
    k_residual<<<NB, NT, 0, stream>>>(x, src, V, partial);
    k_reduce_scalar<<<1, NT, 0, stream>>>(partial, scal, SC_BETA);
    k_scale<<<NB, NT, 0, stream>>>(V, V, scal, SC_INVB);

    for (int j = 0; j < RESTART; ++j) {
      const int jc = j + 1;  // rows 0..j of V are populated; later rows are not read
      k_stencil<<<NB, NT, 0, stream>>>(V + (size_t)j * NN, w);
      // CGS pass 1
      k_dot_partial<<<NB, NT, 0, stream>>>(V, w, jc, partial);
      k_reduce_dots<<<1, NT, 0, stream>>>(partial, jc, h1);
      // w -= V h1, fused with CGS pass 2 dots
      k_sub_fused<1><<<NB, NT, 0, stream>>>(w, V, h1, jc, partial);
      k_reduce_dots<<<1, NT, 0, stream>>>(partial, jc, h2);
      // w -= V h2, fused with ||w||^2
      k_sub_fused<0><<<NB, NT, 0, stream>>>(w, V, h2, jc, partial);
      k_norm_writeH<<<1, NT, 0, stream>>>(partial, h1, h2, Hm, scal, j);
      // V[j+1] = w / (hn + eps)
      k_scale<<<NB, NT, 0, stream>>>(V + (size_t)(j + 1) * NN, w, scal, SC_INVHN);
    }

    // y = (H^T H + eps I)^{-1} (beta H^T e1)   [WMMA Gram + tiny solve]
    k_solve<<<1, 32, 0, stream>>>(Hm, scal, y);
    // x += sum_k y[k] V[k]
    k_update<<<NB, NT, 0, stream>>>(x, V, y);
  }
}